// GNN_60962765800240
// MI455X (gfx1250) — compile-verified
//
#include <hip/hip_runtime.h>
#include <cstdint>

typedef float v2f __attribute__((ext_vector_type(2)));
typedef float v8f __attribute__((ext_vector_type(8)));

#define N_NODES  100000
#define N_EDGES  3200000
#define N_GRAPHS 1024
#define M_TILES  (N_NODES / 16)   // 6250 exactly

// ---------------------------------------------------------------------------
// deg[n] = 1.0 (self loop)
__global__ void init_deg_kernel(float* __restrict__ deg) {
    int n = blockIdx.x * blockDim.x + threadIdx.x;
    if (n < N_NODES) deg[n] = 1.0f;
}

// deg[dst[e]] += 1
__global__ void count_deg_kernel(const int* __restrict__ dst, float* __restrict__ deg) {
    int e = blockIdx.x * blockDim.x + threadIdx.x;
    if (e < N_EDGES) atomicAdd(&deg[dst[e]], 1.0f);
}

// dinv = rsqrt(deg) (in place); agg1[n,:] = x[n,:] * dinv^2  (self-loop term)
__global__ void dinv_self1_kernel(float* __restrict__ deg_dinv,
                                  const float* __restrict__ x,
                                  float* __restrict__ agg1) {
    int n = blockIdx.x * blockDim.x + threadIdx.x;
    if (n >= N_NODES) return;
    float di = rsqrtf(deg_dinv[n]);
    deg_dinv[n] = di;
    float s = di * di;
    const float4* xr = (const float4*)(x + (size_t)n * 16);
    float4*       ar = (float4*)(agg1 + (size_t)n * 16);
#pragma unroll
    for (int i = 0; i < 4; ++i) {
        float4 v = xr[i];
        ar[i] = make_float4(v.x * s, v.y * s, v.z * s, v.w * s);
    }
}

// agg1[dst,:] += x[src,:] * (dinv[src]*dinv[dst])   (16 feats/edge)
__global__ void edge_agg1_kernel(const int* __restrict__ src,
                                 const int* __restrict__ dst,
                                 const float* __restrict__ dinv,
                                 const float* __restrict__ x,
                                 float* __restrict__ agg1) {
    int e = blockIdx.x * blockDim.x + threadIdx.x;
    if (e >= N_EDGES) return;
    int s = src[e], d = dst[e];
    float nrm = dinv[s] * dinv[d];
    const float4* xr = (const float4*)(x + (size_t)s * 16);
    float* out = agg1 + (size_t)d * 16;
#pragma unroll
    for (int i = 0; i < 4; ++i) {
        float4 v = xr[i];
        atomicAdd(out + 4 * i + 0, v.x * nrm);
        atomicAdd(out + 4 * i + 1, v.y * nrm);
        atomicAdd(out + 4 * i + 2, v.z * nrm);
        atomicAdd(out + 4 * i + 3, v.w * nrm);
    }
}

// ---------------------------------------------------------------------------
// Fused:  h1 = relu(agg1 @ W1 + b1)  (16->64, WMMA, staged in LDS)
//         t  = h1 @ W2               (64->32, WMMA)
// One wave per 16-row M-tile. 10 waves/block * 625 blocks = 6250 tiles exact.
__global__ __launch_bounds__(320) void gemm12_kernel(
        const float* __restrict__ agg1, const float* __restrict__ W1,
        const float* __restrict__ b1,   const float* __restrict__ W2,
        float* __restrict__ t) {
    __shared__ float hs[10][16][68];          // 16x64 h1 tile per wave, padded

    const int lane = threadIdx.x & 31;
    const int wv   = threadIdx.x >> 5;
    const int tile = blockIdx.x * 10 + wv;
    const int half = lane >> 4;               // 0: K/M group lo, 1: hi
    const int l    = lane & 15;

    // ---------------- GEMM1: [16x16] @ [16x64] ----------------
    v8f acc[4] = {};
    const int row = tile * 16 + l;
#pragma unroll
    for (int kk = 0; kk < 16; kk += 4) {
        const int k0 = kk + half * 2;
        v2f a;
        a.x = agg1[row * 16 + k0];
        a.y = agg1[row * 16 + k0 + 1];
#pragma unroll
        for (int nt = 0; nt < 4; ++nt) {
            v2f b;
            b.x = W1[k0 * 64 + nt * 16 + l];
            b.y = W1[(k0 + 1) * 64 + nt * 16 + l];
            acc[nt] = __builtin_amdgcn_wmma_f32_16x16x4_f32(
                false, a, false, b, (short)0, acc[nt], false, false);
        }
    }
    // bias + relu -> LDS (C layout: VGPR j = row half*8+j, col = nt*16+l)
#pragma unroll
    for (int nt = 0; nt < 4; ++nt) {
        const int c = nt * 16 + l;
        const float bb = b1[c];
#pragma unroll
        for (int j = 0; j < 8; ++j) {
            float v = acc[nt][j] + bb;
            hs[wv][half * 8 + j][c] = fmaxf(v, 0.0f);
        }
    }
    __syncthreads();

    // ---------------- GEMM2: [16x64] @ [64x32] ----------------
    v8f d0 = {}, d1 = {};
#pragma unroll
    for (int kk = 0; kk < 64; kk += 4) {
        const int k0 = kk + half * 2;
        v2f a;
        a.x = hs[wv][l][k0];
        a.y = hs[wv][l][k0 + 1];
        v2f b0, b2v;
        b0.x  = W2[k0 * 32 + l];       b0.y  = W2[(k0 + 1) * 32 + l];
        b2v.x = W2[k0 * 32 + 16 + l];  b2v.y = W2[(k0 + 1) * 32 + 16 + l];
        d0 = __builtin_amdgcn_wmma_f32_16x16x4_f32(false, a, false, b0,  (short)0, d0, false, false);
        d1 = __builtin_amdgcn_wmma_f32_16x16x4_f32(false, a, false, b2v, (short)0, d1, false, false);
    }
#pragma unroll
    for (int j = 0; j < 8; ++j) {
        const int r = tile * 16 + half * 8 + j;
        t[r * 32 + l]      = d0[j];
        t[r * 32 + 16 + l] = d1[j];
    }
}

// ---------------------------------------------------------------------------
// agg2[n,:] = t[n,:] * dinv[n]^2   (self-loop term, 32 feats)
__global__ void self2_kernel(const float* __restrict__ dinv,
                             const float* __restrict__ t,
                             float* __restrict__ agg2) {
    int n = blockIdx.x * blockDim.x + threadIdx.x;
    if (n >= N_NODES) return;
    float di = dinv[n];
    float s = di * di;
    const float4* tr = (const float4*)(t + (size_t)n * 32);
    float4*       ar = (float4*)(agg2 + (size_t)n * 32);
#pragma unroll
    for (int i = 0; i < 8; ++i) {
        float4 v = tr[i];
        ar[i] = make_float4(v.x * s, v.y * s, v.z * s, v.w * s);
    }
}

// agg2[dst,:] += t[src,:] * norm   (32 feats/edge)
__global__ void edge_agg2_kernel(const int* __restrict__ src,
                                 const int* __restrict__ dst,
                                 const float* __restrict__ dinv,
                                 const float* __restrict__ t,
                                 float* __restrict__ agg2) {
    int e = blockIdx.x * blockDim.x + threadIdx.x;
    if (e >= N_EDGES) return;
    int s = src[e], d = dst[e];
    float nrm = dinv[s] * dinv[d];
    const float4* tr = (const float4*)(t + (size_t)s * 32);
    float* out = agg2 + (size_t)d * 32;
#pragma unroll
    for (int i = 0; i < 8; ++i) {
        float4 v = tr[i];
        atomicAdd(out + 4 * i + 0, v.x * nrm);
        atomicAdd(out + 4 * i + 1, v.y * nrm);
        atomicAdd(out + 4 * i + 2, v.z * nrm);
        atomicAdd(out + 4 * i + 3, v.w * nrm);
    }
}

__global__ void zero_kernel(float* __restrict__ p, int n) {
    int i = blockIdx.x * blockDim.x + threadIdx.x;
    if (i < n) p[i] = 0.0f;
}

// h2 = relu(agg2 + b2); gsum[batch[n],:] += h2; gcnt[batch[n]] += 1
__global__ void relu_pool_kernel(const float* __restrict__ agg2,
                                 const float* __restrict__ b2,
                                 const int* __restrict__ batch,
                                 float* __restrict__ gsum,
                                 float* __restrict__ gcnt) {
    int n = blockIdx.x * blockDim.x + threadIdx.x;
    if (n >= N_NODES) return;
    int g = batch[n];
    atomicAdd(&gcnt[g], 1.0f);
    const float* a = agg2 + (size_t)n * 32;
    float* gs = gsum + (size_t)g * 32;
#pragma unroll
    for (int f = 0; f < 32; ++f) {
        float v = fmaxf(a[f] + b2[f], 0.0f);
        atomicAdd(gs + f, v);
    }
}

// g = gsum/cnt; h = relu(g@fc1W+fc1b); out = sigmoid(h@fc2W+fc2b)
__global__ void mlp_kernel(const float* __restrict__ gsum,
                           const float* __restrict__ gcnt,
                           const float* __restrict__ fc1W,
                           const float* __restrict__ fc1b,
                           const float* __restrict__ fc2W,
                           const float* __restrict__ fc2b,
                           float* __restrict__ out) {
    int g = blockIdx.x * blockDim.x + threadIdx.x;
    if (g >= N_GRAPHS) return;
    float inv = 1.0f / fmaxf(gcnt[g], 1.0f);
    float gv[32];
#pragma unroll
    for (int f = 0; f < 32; ++f) gv[f] = gsum[g * 32 + f] * inv;
    float o = fc2b[0];
#pragma unroll
    for (int j = 0; j < 16; ++j) {
        float s = fc1b[j];
#pragma unroll
        for (int f = 0; f < 32; ++f) s = fmaf(gv[f], fc1W[f * 16 + j], s);
        o = fmaf(fmaxf(s, 0.0f), fc2W[j], o);
    }
    out[g] = 1.0f / (1.0f + expf(-o));
}

// ---------------------------------------------------------------------------
extern "C" void kernel_launch(void* const* d_in, const int* in_sizes, int n_in,
                              void* d_out, int out_size, void* d_ws, size_t ws_size,
                              hipStream_t stream) {
    const float* x    = (const float*)d_in[0];
    const int*   ei   = (const int*)d_in[1];
    const int*   bat  = (const int*)d_in[2];
    const float* W1   = (const float*)d_in[3];
    const float* b1   = (const float*)d_in[4];
    const float* W2   = (const float*)d_in[5];
    const float* b2   = (const float*)d_in[6];
    const float* fc1W = (const float*)d_in[7];
    const float* fc1b = (const float*)d_in[8];
    const float* fc2W = (const float*)d_in[9];
    const float* fc2b = (const float*)d_in[10];
    float* out = (float*)d_out;

    char* ws = (char*)d_ws;
    float* deg  = (float*)(ws + 0);          // 100000 f32 -> becomes dinv
    float* agg1 = (float*)(ws + 400000);     // 100000*16 f32
    float* t    = (float*)(ws + 6800000);    // 100000*32 f32
    float* agg2 = (float*)(ws + 19600000);   // 100000*32 f32
    float* gsum = (float*)(ws + 32400000);   // 1024*32 f32
    float* gcnt = (float*)(ws + 32531072);   // 1024 f32

    const int* src = ei;
    const int* dst = ei + N_EDGES;

    const int NB_N = (N_NODES + 255) / 256;   // 391
    const int NB_E = N_EDGES / 256;           // 12500 exact

    init_deg_kernel<<<NB_N, 256, 0, stream>>>(deg);
    count_deg_kernel<<<NB_E, 256, 0, stream>>>(dst, deg);
    dinv_self1_kernel<<<NB_N, 256, 0, stream>>>(deg, x, agg1);
    edge_agg1_kernel<<<NB_E, 256, 0, stream>>>(src, dst, deg, x, agg1);

    gemm12_kernel<<<M_TILES / 10, 320, 0, stream>>>(agg1, W1, b1, W2, t);

    self2_kernel<<<NB_N, 256, 0, stream>>>(deg, t, agg2);
    edge_agg2_kernel<<<NB_E, 256, 0, stream>>>(src, dst, deg, t, agg2);

    zero_kernel<<<(N_GRAPHS * 32 + 255) / 256, 256, 0, stream>>>(gsum, N_GRAPHS * 32);
    zero_kernel<<<(N_GRAPHS + 255) / 256, 256, 0, stream>>>(gcnt, N_GRAPHS);
    relu_pool_kernel<<<NB_N, 256, 0, stream>>>(agg2, b2, bat, gsum, gcnt);

    mlp_kernel<<<(N_GRAPHS + 255) / 256, 256, 0, stream>>>(
        gsum, gcnt, fc1W, fc1b, fc2W, fc2b, out);
}